// COMPoolNet_h_89060441850432
// MI455X (gfx1250) — compile-verified
//
#include <hip/hip_runtime.h>
#include <hip/hip_bf16.h>

typedef float v2f __attribute__((ext_vector_type(2)));
typedef float v8f __attribute__((ext_vector_type(8)));

#define GB   256     // graphs per batch
#define NP   256     // nodes per graph (stage 1)
#define FD   128     // feature dim
#define NC   10      // classes
#define EPG  4096    // edges per graph
#define NE   (GB*EPG)

// ---------------------------------------------------------------------------
// utility: zero a float buffer
// ---------------------------------------------------------------------------
__global__ void k_zero(float* __restrict__ p, int n) {
    int i = blockIdx.x * blockDim.x + threadIdx.x;
    if (i < n) p[i] = 0.f;
}

// ---------------------------------------------------------------------------
// Per-graph GCN aggregation, fully LDS-resident (needs n_per*FD*4 + 2*n_per*4
// bytes of dynamic LDS; stage 1 = 130 KB, well inside the 320 KB WGP LDS).
// Computes deg_out/deg_in and agg = Dn^-1/2 * A * Dd^-1/2 * x per graph.
// Edges with em<=0 are skipped exactly (they contribute 0 in the reference,
// and their node ids may point outside this graph).
// ---------------------------------------------------------------------------
__global__ __launch_bounds__(256)
void k_gcn_agg(const float* __restrict__ x,
               const int* __restrict__ src, const int* __restrict__ dst,
               const float* __restrict__ em,
               float* __restrict__ agg,
               float* __restrict__ degO, float* __restrict__ degI,
               int n_per)
{
    extern __shared__ float lds[];
    float* aggL = lds;                 // n_per * FD
    float* doL  = lds + n_per * FD;    // n_per
    float* diL  = doL + n_per;         // n_per
    const int b = blockIdx.x, node0 = b * n_per, e0 = b * EPG;
    const int tid = threadIdx.x;

    for (int i = tid; i < n_per * FD; i += 256) aggL[i] = 0.f;
    for (int i = tid; i < n_per;      i += 256) { doL[i] = 0.f; diL[i] = 0.f; }
    __syncthreads();

    // pass 1: degrees (LDS float atomics -> ds_add_f32)
    for (int e = tid; e < EPG; e += 256) {
        float m = em ? em[e0 + e] : 1.0f;
        if (m > 0.f) {
            atomicAdd(&doL[src[e0 + e] - node0], 1.0f);
            atomicAdd(&diL[dst[e0 + e] - node0], 1.0f);
        }
    }
    __syncthreads();

    // pass 2: scatter-add normalized source features (32 lanes x float4 per edge)
    const int lane = tid & 31;   // column chunk: cols [4*lane, 4*lane+3]
    const int grp  = tid >> 5;   // 8 edges in flight per iteration
    for (int e = grp; e < EPG; e += 8) {
        float m = em ? em[e0 + e] : 1.0f;
        if (m <= 0.f) continue;
        const int ls = src[e0 + e] - node0;
        const int ld = dst[e0 + e] - node0;
        const float s = rsqrtf(fmaxf(doL[ls], 1.0f));
        const float4 v = *(const float4*)(x + (size_t)(node0 + ls) * FD + lane * 4);
        float* a = aggL + ld * FD + lane * 4;
        atomicAdd(a + 0, v.x * s); atomicAdd(a + 1, v.y * s);
        atomicAdd(a + 2, v.z * s); atomicAdd(a + 3, v.w * s);
    }
    __syncthreads();

    // pass 3: apply in-degree norm, spill agg + degrees
    for (int i = tid; i < n_per * FD; i += 256) {
        const int n = i / FD;
        agg[(size_t)node0 * FD + i] = aggL[i] * rsqrtf(fmaxf(diL[n], 1.0f));
    }
    for (int i = tid; i < n_per; i += 256) {
        degO[node0 + i] = doL[i];
        degI[node0 + i] = diL[i];
    }
}

// ---------------------------------------------------------------------------
// WMMA f32 GEMM: out = act(A[MxK] @ W[KxN] + bias) (+resid).
// M%16==0, N%16==0, K%4==0. Block = 256 threads (8 waves); wave w owns the
// 16x16 output tile at columns [16*(8*by+w), ...). A 16xK strip is staged in
// LDS. v_wmma_f32_16x16x4_f32 keeps full f32 precision.
// A layout (ISA 7.12.2): lane<16 -> {A[l][k0],A[l][k0+1]}, lane>=16 -> k0+2/3.
// C/D layout: VGPR r -> row r (lanes 0-15) / row r+8 (lanes 16-31).
// ---------------------------------------------------------------------------
__global__ __launch_bounds__(256)
void k_gemm_wmma(const float* __restrict__ A, const float* __restrict__ W,
                 const float* __restrict__ bias, const float* __restrict__ resid,
                 float* __restrict__ out, int M, int N, int K, int relu)
{
    extern __shared__ float AL[];            // 16 x K
    const int row0 = blockIdx.x * 16;
    const int tid  = threadIdx.x;
    const int wave = tid >> 5;
    const int lane = tid & 31;

    for (int i = tid; i < 16 * K; i += 256)
        AL[i] = A[(size_t)(row0 + i / K) * K + (i % K)];
    __syncthreads();

    const int col0 = (blockIdx.y * 8 + wave) * 16;
    if (col0 >= N) return;

    const int half = lane >> 4;              // 0: lanes 0-15, 1: lanes 16-31
    const int l16  = lane & 15;
    v8f acc = {0.f, 0.f, 0.f, 0.f, 0.f, 0.f, 0.f, 0.f};

    for (int k0 = 0; k0 < K; k0 += 4) {
        v2f a, bm;
        a.x  = AL[l16 * K + k0 + 2 * half + 0];
        a.y  = AL[l16 * K + k0 + 2 * half + 1];
        bm.x = W[(size_t)(k0 + 2 * half + 0) * N + col0 + l16];
        bm.y = W[(size_t)(k0 + 2 * half + 1) * N + col0 + l16];
        acc = __builtin_amdgcn_wmma_f32_16x16x4_f32(
                  false, a, false, bm, (short)0, acc, false, false);
    }

    #pragma unroll
    for (int r = 0; r < 8; ++r) {
        const int mrow = r + 8 * half;
        const int col  = col0 + l16;
        float v = acc[r] + bias[col];
        if (relu)  v = fmaxf(v, 0.f);
        if (resid) v += resid[(size_t)(row0 + mrow) * N + col];   // x + relu(conv)
        out[(size_t)(row0 + mrow) * N + col] = v;
    }
}

// ---------------------------------------------------------------------------
// Pooling score: score = Din^-1/2 * A * (Dout^-1/2 * (x @ Ws)) + bs.
// Projection to scalar done BEFORE the edge scatter (linear-op reorder):
// 128x less edge traffic than the literal reference order.
// ---------------------------------------------------------------------------
__global__ __launch_bounds__(256)
void k_score(const float* __restrict__ x,
             const int* __restrict__ src, const int* __restrict__ dst,
             const float* __restrict__ em,
             const float* __restrict__ degO, const float* __restrict__ degI,
             const float* __restrict__ Ws, const float* __restrict__ bs,
             float* __restrict__ score, int n_per)
{
    __shared__ float p[256];
    __shared__ float sacc[256];
    const int b = blockIdx.x, node0 = b * n_per, e0 = b * EPG;
    const int tid = threadIdx.x;

    for (int n = tid; n < n_per; n += 256) {
        const float* xr = x + (size_t)(node0 + n) * FD;
        float acc = 0.f;
        for (int c = 0; c < FD; ++c) acc += xr[c] * Ws[c];
        p[n]    = acc * rsqrtf(fmaxf(degO[node0 + n], 1.f));
        sacc[n] = 0.f;
    }
    __syncthreads();
    for (int e = tid; e < EPG; e += 256) {
        float m = em ? em[e0 + e] : 1.0f;
        if (m > 0.f)
            atomicAdd(&sacc[dst[e0 + e] - node0], p[src[e0 + e] - node0]);
    }
    __syncthreads();
    for (int n = tid; n < n_per; n += 256)
        score[node0 + n] = sacc[n] * rsqrtf(fmaxf(degI[node0 + n], 1.f)) + bs[0];
}

// ---------------------------------------------------------------------------
// Per-graph top-k pooling: exact rank (top_k tie-break: value desc, idx asc),
// tanh-gated feature split into dis/com, edge remap into new dense ids.
// ---------------------------------------------------------------------------
__global__ __launch_bounds__(256)
void k_pool(const float* __restrict__ x, const float* __restrict__ score,
            const int* __restrict__ src, const int* __restrict__ dst,
            const float* __restrict__ em,
            float* __restrict__ fdis, float* __restrict__ fcom,
            int* __restrict__ srcO, int* __restrict__ dstO, float* __restrict__ emO,
            int n_per, int k, int do_edges)
{
    __shared__ float s[256];
    __shared__ int   kept[256];
    __shared__ int   pos[256];
    __shared__ float g[256];
    const int b = blockIdx.x, node0 = b * n_per, e0 = b * EPG;
    const int tid = threadIdx.x;

    for (int n = tid; n < n_per; n += 256) s[n] = score[node0 + n];
    __syncthreads();
    for (int n = tid; n < n_per; n += 256) {
        const float sn = s[n];
        int rank = 0;
        for (int j = 0; j < n_per; ++j) {
            const float sj = s[j];
            rank += (sj > sn) || (sj == sn && j < n);
        }
        kept[n] = (rank < k);
        g[n]    = tanhf(sn);
    }
    __syncthreads();
    for (int n = tid; n < n_per; n += 256) {
        int cnt = 0;
        if (kept[n]) { for (int j = 0; j < n; ++j) cnt += kept[j];      pos[n] = b * k + cnt; }
        else         { for (int j = 0; j < n; ++j) cnt += 1 - kept[j];  pos[n] = b * (n_per - k) + cnt; }
    }
    __syncthreads();

    // gated feature split
    for (int i = tid; i < n_per * FD; i += 256) {
        const int n = i / FD, c = i % FD;
        const float v = x[(size_t)(node0 + n) * FD + c] * g[n];
        if (kept[n]) fdis[(size_t)pos[n] * FD + c] = v;
        else         fcom[(size_t)pos[n] * FD + c] = v;
    }

    if (do_edges) {
        for (int e = tid; e < EPG; e += 256) {
            float m = em ? em[e0 + e] : 1.0f;
            int so = 0, dro = 0; float mo = 0.f;
            if (m > 0.f) {
                const int ls = src[e0 + e] - node0;
                const int ld = dst[e0 + e] - node0;
                if (kept[ls] && kept[ld]) { mo = 1.f; so = pos[ls]; dro = pos[ld]; }
            }
            srcO[e0 + e] = so; dstO[e0 + e] = dro; emO[e0 + e] = mo;
        }
    }
}

// ---------------------------------------------------------------------------
// Readout: accum[b, 0:D] += mean, accum[b, D:2D] += max (accum pre-zeroed).
// Optional copy-out (for hg3_com). One block/graph, thread = column.
// ---------------------------------------------------------------------------
__global__ __launch_bounds__(128)
void k_readout(const float* __restrict__ f, float* __restrict__ accum,
               float* __restrict__ copy, int rows)
{
    const int b = blockIdx.x, c = threadIdx.x;
    const float* fb = f + (size_t)b * rows * FD + c;
    float sum = 0.f, mx = -INFINITY;
    for (int r = 0; r < rows; ++r) {
        const float v = fb[(size_t)r * FD];
        sum += v; mx = fmaxf(mx, v);
    }
    const float mean = sum / (float)rows;
    accum[b * 2 * FD + c]      += mean;
    accum[b * 2 * FD + FD + c] += mx;
    if (copy) { copy[b * 2 * FD + c] = mean; copy[b * 2 * FD + FD + c] = mx; }
}

// ---------------------------------------------------------------------------
// Global softmax over the flat stage-1 score vector (single 1024-thread block)
// ---------------------------------------------------------------------------
__global__ __launch_bounds__(1024)
void k_softmax(const float* __restrict__ s, float* __restrict__ out, int n)
{
    __shared__ float red[1024];
    const int tid = threadIdx.x;
    float mx = -INFINITY;
    for (int i = tid; i < n; i += 1024) mx = fmaxf(mx, s[i]);
    red[tid] = mx; __syncthreads();
    for (int st = 512; st > 0; st >>= 1) {
        if (tid < st) red[tid] = fmaxf(red[tid], red[tid + st]);
        __syncthreads();
    }
    mx = red[0]; __syncthreads();
    float sum = 0.f;
    for (int i = tid; i < n; i += 1024) sum += expf(s[i] - mx);
    red[tid] = sum; __syncthreads();
    for (int st = 512; st > 0; st >>= 1) {
        if (tid < st) red[tid] += red[tid + st];
        __syncthreads();
    }
    const float inv = 1.0f / red[0];
    for (int i = tid; i < n; i += 1024) out[i] = expf(s[i] - mx) * inv;
}

// ---------------------------------------------------------------------------
// Scalar linear layer for non-16-multiple dims (final N=10 heads)
// ---------------------------------------------------------------------------
__global__ void k_linear(const float* __restrict__ in, const float* __restrict__ W,
                         const float* __restrict__ bias, float* __restrict__ out,
                         int M, int N, int K, int relu)
{
    const int i = blockIdx.x * blockDim.x + threadIdx.x;
    if (i >= M * N) return;
    const int m = i / N, n = i % N;
    float acc = bias[n];
    const float* a = in + (size_t)m * K;
    for (int kk = 0; kk < K; ++kk) acc += a[kk] * W[(size_t)kk * N + n];
    out[i] = relu ? fmaxf(acc, 0.f) : acc;
}

// ---------------------------------------------------------------------------
extern "C" void kernel_launch(void* const* d_in, const int* in_sizes, int n_in,
                              void* d_out, int out_size, void* d_ws, size_t ws_size,
                              hipStream_t stream) {
    (void)in_sizes; (void)n_in; (void)out_size; (void)ws_size;
    const float* feature = (const float*)d_in[0];
    const int*   src     = (const int*)  d_in[1];
    const int*   dst     = (const int*)  d_in[2];
    // d_in[3] = label (unused by forward outputs)
    const float* W1 = (const float*)d_in[4],  *b1 = (const float*)d_in[5];
    const float* W2 = (const float*)d_in[6],  *b2 = (const float*)d_in[7];
    const float* W3 = (const float*)d_in[8],  *b3 = (const float*)d_in[9];
    const float* Ws = (const float*)d_in[10], *bs = (const float*)d_in[11];
    const float* Wg0= (const float*)d_in[12], *bg0= (const float*)d_in[13];
    const float* Wg1= (const float*)d_in[14], *bg1= (const float*)d_in[15];
    const float* Wg2= (const float*)d_in[16], *bg2= (const float*)d_in[17];
    const float* Wp0= (const float*)d_in[18], *bp0= (const float*)d_in[19];
    const float* Wp1= (const float*)d_in[20], *bp1= (const float*)d_in[21];
    const float* Wp2= (const float*)d_in[22], *bp2= (const float*)d_in[23];

    // ---- workspace carve ----
    float* wsf = (float*)d_ws;  size_t o = 0;
    auto alloc = [&](size_t n) { float* p = wsf + o; o += n; return p; };
    float* agg  = alloc((size_t)GB*NP*FD);      // 65536x128, reused each stage
    float* out1 = alloc((size_t)GB*NP*FD);
    float* f1d  = alloc((size_t)GB*128*FD);
    float* fcom = alloc((size_t)GB*128*FD);     // com buffer, reused each stage
    float* out2 = alloc((size_t)GB*128*FD);
    float* f2d  = alloc((size_t)GB*64*FD);
    float* out3 = alloc((size_t)GB*64*FD);
    float* f3d  = alloc((size_t)GB*32*FD);
    float* degO = alloc((size_t)GB*NP);
    float* degI = alloc((size_t)GB*NP);
    float* score= alloc((size_t)GB*NP);
    int*   s1   = (int*)alloc(NE);
    int*   d1   = (int*)alloc(NE);
    float* em1  = alloc(NE);
    int*   s2   = (int*)alloc(NE);
    int*   d2   = (int*)alloc(NE);
    float* em2  = alloc(NE);
    float* hg   = alloc((size_t)GB*2*FD);
    float* hgc  = alloc((size_t)GB*2*FD);       // contiguous with hg
    float* t0   = alloc((size_t)GB*64*64);      // mlp hidden (max 16384x64)
    float* t1   = alloc((size_t)GB*64*32);
    float* t2   = alloc((size_t)GB*FD);
    float* t3   = alloc((size_t)GB*64);

    // ---- output carve (return order) ----
    float* o_scores  = (float*)d_out;                 // [256,10]
    float* o_scoresc = o_scores  + GB*NC;             // [256,10]
    float* o_hg3c    = o_scoresc + GB*NC;             // [256,256]
    float* o_npred   = o_hg3c    + GB*2*FD;           // [16384,10]
    float* o_ns1     = o_npred   + (size_t)GB*64*NC;  // [65536]

    // zero hg/hg_com accumulators (contiguous)
    k_zero<<<(GB*4*FD + 255)/256, 256, 0, stream>>>(hg, GB*4*FD);

    const dim3 blk(256);

    // ================= stage 1 (n_per=256 -> k=128) =================
    k_gcn_agg<<<GB, blk, (NP*FD + 2*NP)*sizeof(float), stream>>>(
        feature, src, dst, nullptr, agg, degO, degI, NP);
    k_gemm_wmma<<<dim3(GB*NP/16, 1), blk, 16*FD*sizeof(float), stream>>>(
        agg, W1, b1, feature, out1, GB*NP, FD, FD, 1);
    k_score<<<GB, blk, 0, stream>>>(out1, src, dst, nullptr, degO, degI, Ws, bs, score, NP);
    k_softmax<<<1, 1024, 0, stream>>>(score, o_ns1, GB*NP);
    k_pool<<<GB, blk, 0, stream>>>(out1, score, src, dst, nullptr,
                                   f1d, fcom, s1, d1, em1, NP, 128, 1);
    k_readout<<<GB, 128, 0, stream>>>(f1d,  hg,  nullptr, 128);
    k_readout<<<GB, 128, 0, stream>>>(fcom, hgc, nullptr, 128);

    // ================= stage 2 (n_per=128 -> k=64) =================
    k_gcn_agg<<<GB, blk, (128*FD + 2*128)*sizeof(float), stream>>>(
        f1d, s1, d1, em1, agg, degO, degI, 128);
    k_gemm_wmma<<<dim3(GB*128/16, 1), blk, 16*FD*sizeof(float), stream>>>(
        agg, W2, b2, f1d, out2, GB*128, FD, FD, 1);
    k_score<<<GB, blk, 0, stream>>>(out2, s1, d1, em1, degO, degI, Ws, bs, score, 128);
    k_pool<<<GB, blk, 0, stream>>>(out2, score, s1, d1, em1,
                                   f2d, fcom, s2, d2, em2, 128, 64, 1);
    k_readout<<<GB, 128, 0, stream>>>(f2d,  hg,  nullptr, 64);
    k_readout<<<GB, 128, 0, stream>>>(fcom, hgc, nullptr, 64);

    // ================= stage 3 (n_per=64 -> k=32) =================
    k_gcn_agg<<<GB, blk, (64*FD + 2*64)*sizeof(float), stream>>>(
        f2d, s2, d2, em2, agg, degO, degI, 64);
    k_gemm_wmma<<<dim3(GB*64/16, 1), blk, 16*FD*sizeof(float), stream>>>(
        agg, W3, b3, f2d, out3, GB*64, FD, FD, 1);
    k_score<<<GB, blk, 0, stream>>>(out3, s2, d2, em2, degO, degI, Ws, bs, score, 64);
    k_pool<<<GB, blk, 0, stream>>>(out3, score, s2, d2, em2,
                                   f3d, fcom, nullptr, nullptr, nullptr, 64, 32, 0);
    k_readout<<<GB, 128, 0, stream>>>(f3d,  hg,  nullptr, 32);
    k_readout<<<GB, 128, 0, stream>>>(fcom, hgc, o_hg3c, 32);   // hg3_com copy-out

    // ================= heads =================
    // node_pred = mlp(out3): 128->64 relu, 64->32 relu, 32->10
    k_gemm_wmma<<<dim3(GB*64/16, 1), blk, 16*FD*sizeof(float), stream>>>(
        out3, Wp0, bp0, nullptr, t0, GB*64, 64, FD, 1);
    k_gemm_wmma<<<dim3(GB*64/16, 1), blk, 16*64*sizeof(float), stream>>>(
        t0, Wp1, bp1, nullptr, t1, GB*64, 32, 64, 1);
    k_linear<<<(GB*64*NC + 255)/256, 256, 0, stream>>>(t1, Wp2, bp2, o_npred, GB*64, NC, 32, 0);

    // scores = mlp(hg): 256->128 relu, 128->64 relu, 64->10
    k_gemm_wmma<<<dim3(GB/16, 1), blk, 16*2*FD*sizeof(float), stream>>>(
        hg, Wg0, bg0, nullptr, t2, GB, FD, 2*FD, 1);
    k_gemm_wmma<<<dim3(GB/16, 1), blk, 16*FD*sizeof(float), stream>>>(
        t2, Wg1, bg1, nullptr, t3, GB, 64, FD, 1);
    k_linear<<<(GB*NC + 255)/256, 256, 0, stream>>>(t3, Wg2, bg2, o_scores, GB, NC, 64, 0);

    // scores_com = mlp(hg_com)
    k_gemm_wmma<<<dim3(GB/16, 1), blk, 16*2*FD*sizeof(float), stream>>>(
        hgc, Wg0, bg0, nullptr, t2, GB, FD, 2*FD, 1);
    k_gemm_wmma<<<dim3(GB/16, 1), blk, 16*FD*sizeof(float), stream>>>(
        t2, Wg1, bg1, nullptr, t3, GB, 64, FD, 1);
    k_linear<<<(GB*NC + 255)/256, 256, 0, stream>>>(t3, Wg2, bg2, o_scoresc, GB, NC, 64, 0);
}